// Attention_49967649522049
// MI455X (gfx1250) — compile-verified
//
#include <hip/hip_runtime.h>
#include <hip/hip_bf16.h>
#include <cstdint>
#include <cstddef>

// CDNA5 WMMA fragment types for V_WMMA_F32_16X16X4_F32:
//   A: 16x4 f32 -> 2 f32/lane ; B: 4x16 f32 -> 2 f32/lane ; C/D: 16x16 f32 -> 8 f32/lane
typedef __attribute__((ext_vector_type(2))) float v2f;
typedef __attribute__((ext_vector_type(8))) float v8f;

// Tensor DMA descriptor groups (CDNA5 ISA)
typedef unsigned int u32x4 __attribute__((ext_vector_type(4)));
typedef int          i32x4 __attribute__((ext_vector_type(4)));
typedef int          i32x8 __attribute__((ext_vector_type(8)));

#define B_ 2
#define M_ 8
#define I_ 256
#define J_ 64
#define E_ 256
#define H_ 8
#define D_ 32
#define KV_W   (2 * E_)   // 512 combined K|V output columns
#define MEM_LD (E_ + 4)   // 260-dword LDS row stride (TDM pad: 4 dwords / 256 dwords)

__global__ __launch_bounds__(256, 1)
void attn_fused_kernel(const float* __restrict__ x,
                       const float* __restrict__ mems,
                       const uint8_t* __restrict__ mask,
                       const float* __restrict__ Wq,
                       const float* __restrict__ Wkv,
                       const float* __restrict__ Wo,
                       const float* __restrict__ bo,
                       float* __restrict__ out)
{
    // LDS: 64x260 f32 staged mems tile (TDM dest, row-padded) + 64x512 f32 KV.
    __shared__ float s_mem[J_ * MEM_LD];  //  65 KB
    __shared__ float s_kv[J_ * KV_W];     // 128 KB : K cols [0,256), V cols [256,512)
    __shared__ float s_x[E_];
    __shared__ float s_q[E_];             // q, pre-scaled, layout [h*32 + d]
    __shared__ float s_sim[H_ * J_];
    __shared__ float s_o[E_];

    const int tid  = threadIdx.x;
    const int lane = tid & 31;
    const int wave = tid >> 5;
    const int bmi  = blockIdx.x;          // flattened (b, m, i), 0..4095

    const float*   x_row = x    + (size_t)bmi * E_;
    const float*   mem_t = mems + (size_t)bmi * (J_ * E_);
    const uint8_t* msk   = mask + (size_t)bmi * J_;

    // ---- TDM: async DMA of the 64x256 f32 mems tile -> LDS, row-padded ----
    // One tensor_load_to_lds per workgroup (wave 0 issues; TENSORcnt is per-wave).
    if (wave == 0) {
        const uint64_t ga      = (uint64_t)(uintptr_t)mem_t;             // tile start
        const uint32_t lds_adr = (uint32_t)(uintptr_t)(void*)&s_mem[0];  // LDS byte addr

        // D# group 0: count=1(valid), lds_addr[63:32], global_addr[120:64], type=2
        u32x4 g0;
        g0.x = 1u;
        g0.y = lds_adr;
        g0.z = (uint32_t)(ga & 0xFFFFFFFFu);
        g0.w = (uint32_t)((ga >> 32) & 0x01FFFFFFu) | (2u << 30);

        // D# group 1: data_size=4B(2), pad_enable, pad_interval=256dw(7),
        //             pad_amount=4dw(3); tensor 256x64, tile 256x64,
        //             dim0_stride=256, dim1_stride=16384 (elements)
        i32x8 g1;
        g1[0] = (int)((2u << 16) | (1u << 20) | (7u << 22) | (3u << 25));
        g1[1] = (int)((uint32_t)E_ << 16);          // tensor_dim0[15:0] @ bit48
        g1[2] = (int)((uint32_t)J_ << 16);          // tensor_dim0 hi=0 | tensor_dim1 lo
        g1[3] = (int)((uint32_t)E_ << 16);          // tensor_dim1 hi=0 | tile_dim0=256
        g1[4] = (int)((uint32_t)J_);                // tile_dim1=64 | tile_dim2=0
        g1[5] = (int)((uint32_t)E_);                // tensor_dim0_stride lo = 256
        g1[6] = (int)((uint32_t)(J_ * E_) << 16);   // stride0 hi=0 | dim1_stride lo
        g1[7] = 0;                                  // dim1_stride hi = 0

        i32x4 z4 = {0, 0, 0, 0};
#if defined(__clang_major__) && (__clang_major__ >= 23)
        i32x8 z8 = {0, 0, 0, 0, 0, 0, 0, 0};
        __builtin_amdgcn_tensor_load_to_lds(g0, g1, z4, z4, z8, 0);
#else
        __builtin_amdgcn_tensor_load_to_lds(g0, g1, z4, z4, 0);
#endif
    }

    // ---- stage x row (overlaps the TDM DMA) ----
    s_x[tid] = x_row[tid];
    __syncthreads();

    // ---- Q projection: q = (x @ Wq) * d^-0.5 ; overlaps the TDM DMA ----
    {
        const float scale = 0.17677669529663687f; // 1/sqrt(32)
        float acc = 0.0f;
        #pragma unroll 4
        for (int k = 0; k < E_; ++k)
            acc += s_x[k] * Wq[k * E_ + tid];     // coalesced across tid
        s_q[tid] = acc * scale;
    }

    // ---- complete the DMA before consuming s_mem ----
    if (wave == 0)
        __builtin_amdgcn_s_wait_tensorcnt(0);
    __syncthreads();

    // ---- KV projection: [64 x 256] @ [256 x 512] with V_WMMA_F32_16X16X4_F32 ----
    // 128 output tiles of 16x16 (4 row-tiles x 32 col-tiles); each wave owns 16.
    {
        const int lm = lane & 15;          // M (or N) index within tile
        const int lk = (lane >> 4) << 1;   // K sub-offset: lanes 0-15 -> 0, 16-31 -> 2

        for (int tt = 0; tt < 16; ++tt) {
            const int t  = wave * 16 + tt;
            const int m0 = (t >> 5) << 4;          // row-tile base (J dim)
            const int n0 = (t & 31) << 4;          // col-tile base (2E dim)

            v8f c = {};
            // A[m][k] from padded LDS tile: lane = m%16 + 16*(k/2), vgpr j = k%2
            const float* arow = &s_mem[(m0 + lm) * MEM_LD + lk];
            // B[k][n] from L2-resident Wkv
            const float* bcol = Wkv + (size_t)lk * KV_W + (n0 + lm);

            #pragma unroll 4
            for (int kc = 0; kc < E_ / 4; ++kc) {
                const int k0 = kc << 2;
                v2f a = *(const v2f*)(arow + k0);          // ds_load_b64, bank-spread
                v2f b;
                b.x = bcol[(size_t)k0 * KV_W];
                b.y = bcol[(size_t)(k0 + 1) * KV_W];
                c = __builtin_amdgcn_wmma_f32_16x16x4_f32(
                        /*neg_a=*/false, a, /*neg_b=*/false, b,
                        /*c_mod=*/(short)0, c, /*reuse_a=*/false, /*reuse_b=*/false);
            }

            // C layout: VGPR v -> row m0+v (lanes 0-15) / m0+8+v (lanes 16-31), col n0+lm
            const int row0 = m0 + ((lane >> 4) << 3);
            #pragma unroll
            for (int v = 0; v < 8; ++v)
                s_kv[(row0 + v) * KV_W + n0 + lm] = c[v];
        }
    }
    __syncthreads();

    // ---- scores: sim[h][j] = sum_d q[h,d] * K[j][h*32+d]  (+ mask) ----
    for (int idx = tid; idx < H_ * J_; idx += 256) {
        const int h = idx >> 6;
        const int j = idx & (J_ - 1);
        const float* qh = s_q  + h * D_;
        const float* kr = s_kv + j * KV_W + h * D_;
        float acc = 0.0f;
        #pragma unroll
        for (int d = 0; d < D_; ++d)
            acc += qh[d] * kr[d];
        s_sim[idx] = msk[j] ? acc : -3.402823466e38f;
    }
    __syncthreads();

    // ---- softmax over j, one head per thread (8 active, trivial cost) ----
    if (tid < H_) {
        float* row = s_sim + tid * J_;
        float mx = -3.402823466e38f;
        for (int j = 0; j < J_; ++j) mx = fmaxf(mx, row[j]);
        float sum = 0.0f;
        for (int j = 0; j < J_; ++j) { float e = __expf(row[j] - mx); row[j] = e; sum += e; }
        const float inv = 1.0f / sum;
        for (int j = 0; j < J_; ++j) row[j] *= inv;
    }
    __syncthreads();

    // ---- AV: o[h*32+d] = sum_j attn[h][j] * V[j][h*32+d] ----
    {
        const int e = tid;
        const float* ar = s_sim + (e >> 5) * J_;
        float acc = 0.0f;
        #pragma unroll 4
        for (int j = 0; j < J_; ++j)
            acc += ar[j] * s_kv[j * KV_W + E_ + e];
        s_o[e] = acc;
    }
    __syncthreads();

    // ---- output projection: out = o @ Wo + bo ----
    {
        __builtin_prefetch(&Wo[tid], 0, 0);   // gfx1250 global_prefetch_b8
        float acc = bo[tid];
        #pragma unroll 4
        for (int k = 0; k < E_; ++k)
            acc += s_o[k] * Wo[k * E_ + tid]; // coalesced across tid
        out[(size_t)bmi * E_ + tid] = acc;
    }
}

extern "C" void kernel_launch(void* const* d_in, const int* in_sizes, int n_in,
                              void* d_out, int out_size, void* d_ws, size_t ws_size,
                              hipStream_t stream) {
    (void)in_sizes; (void)n_in; (void)out_size; (void)d_ws; (void)ws_size;
    const float*   x    = (const float*)d_in[0];
    const float*   mems = (const float*)d_in[1];
    const uint8_t* mask = (const uint8_t*)d_in[2];
    const float*   Wq   = (const float*)d_in[3];
    const float*   Wkv  = (const float*)d_in[4];
    const float*   Wo   = (const float*)d_in[5];
    const float*   bo   = (const float*)d_in[6];
    float* out = (float*)d_out;

    dim3 grid(B_ * M_ * I_);   // 4096 workgroups, one per (b,m,i)
    dim3 block(256);           // 8 wave32s
    attn_fused_kernel<<<grid, block, 0, stream>>>(x, mems, mask, Wq, Wkv, Wo, bo, out);
}